// keypoint_encoder_36524401885208
// MI455X (gfx1250) — compile-verified
//
#include <hip/hip_runtime.h>
#include <hip/hip_bf16.h>

typedef __attribute__((ext_vector_type(2))) float v2f;
typedef __attribute__((ext_vector_type(8))) float v8f;

#define BQ 4          // batch
#define N0 20000      // input points

// ---------------------------------------------------------------- transpose input -> xyz [B][N][3]
__global__ void k_xyz0(const float* __restrict__ in, float* __restrict__ xyz, int n) {
  int i = blockIdx.x * blockDim.x + threadIdx.x;
  if (i >= n) return;                      // n = B*N0*3
  int k = i % 3, g = i / 3, nn = g % N0, b = g / N0;
  xyz[i] = in[((size_t)b * 6 + k) * N0 + nn];
}

// ---------------------------------------------------------------- pack weights (pad K) + fold BN
__global__ void k_pack_w(const float* __restrict__ W, const float* __restrict__ bias,
                         const float* __restrict__ gamma, const float* __restrict__ beta,
                         float* __restrict__ Wp, float* __restrict__ sc, float* __restrict__ sh,
                         int cout, int cin, int kpad) {
  int i = blockIdx.x * blockDim.x + threadIdx.x;
  if (i < cout * kpad) {
    int o = i / kpad, c = i - o * kpad;
    Wp[i] = (c < cin) ? W[o * cin + c] : 0.f;
  }
  if (i < cout) {
    const float inv = 0.99999500003749972f;     // 1/sqrt(1+1e-5)
    sc[i] = gamma[i] * inv;
    sh[i] = bias[i] * gamma[i] * inv + beta[i];
  }
}

// ---------------------------------------------------------------- farthest point sampling
#define FPS_T 1024
#define FPS_PP 20
__global__ void __launch_bounds__(FPS_T)
k_fps(const float* __restrict__ xyz, int N, int S, int* __restrict__ inds) {
  const int b = blockIdx.x;
  const float* P = xyz + (size_t)b * N * 3;
  const int tid = threadIdx.x;
  const int ppt = (N + FPS_T - 1) / FPS_T;
  float px[FPS_PP], py[FPS_PP], pz[FPS_PP], md[FPS_PP];
  for (int i = 0; i < FPS_PP; ++i) md[i] = 1e10f;
  for (int i = 0; i < ppt; ++i) {
    int n = tid + i * FPS_T;
    if (n < N) { px[i] = P[n*3]; py[i] = P[n*3+1]; pz[i] = P[n*3+2]; }
  }
  __shared__ float s_v[FPS_T / 32];
  __shared__ int   s_i[FPS_T / 32];
  __shared__ int   s_far;
  int far = 0;
  for (int s = 0; s < S; ++s) {
    if (tid == 0) inds[(size_t)b * S + s] = far;
    const float cx = P[far*3], cy = P[far*3+1], cz = P[far*3+2];
    float bestv = -1.f; int besti = 0x7fffffff;
    for (int i = 0; i < ppt; ++i) {
      int n = tid + i * FPS_T;
      if (n < N) {
        float dx = px[i]-cx, dy = py[i]-cy, dz = pz[i]-cz;
        float m = fminf(md[i], dx*dx + dy*dy + dz*dz);
        md[i] = m;
        if (m > bestv) { bestv = m; besti = n; }
      }
    }
    for (int off = 16; off > 0; off >>= 1) {          // wave32 argmax (tie -> lower index)
      float ov = __shfl_xor(bestv, off);
      int   oi = __shfl_xor(besti, off);
      if (ov > bestv || (ov == bestv && oi < besti)) { bestv = ov; besti = oi; }
    }
    if ((tid & 31) == 0) { s_v[tid >> 5] = bestv; s_i[tid >> 5] = besti; }
    __syncthreads();
    if (tid == 0) {
      float bv = s_v[0]; int bi = s_i[0];
      for (int w = 1; w < FPS_T / 32; ++w)
        if (s_v[w] > bv || (s_v[w] == bv && s_i[w] < bi)) { bv = s_v[w]; bi = s_i[w]; }
      s_far = bi;
    }
    __syncthreads();
    far = s_far;
  }
}

// ---------------------------------------------------------------- gather sampled centers
__global__ void k_gather_xyz(const float* __restrict__ xyz, const int* __restrict__ inds,
                             float* __restrict__ out, int N, int S, int n /* B*S*3 */) {
  int i = blockIdx.x * blockDim.x + threadIdx.x;
  if (i >= n) return;
  int k = i % 3, g = i / 3, b = g / S;
  out[i] = xyz[((size_t)b * N + inds[g]) * 3 + k];
}

// ---------------------------------------------------------------- ball query: one wave per center
__global__ void k_ballquery(const float* __restrict__ xyz, const float* __restrict__ nxyz,
                            int N, int S, int NSAMP, float r2, int* __restrict__ idx, int BS) {
  int gw = (blockIdx.x * blockDim.x + threadIdx.x) >> 5;
  if (gw >= BS) return;
  int lane = threadIdx.x & 31;
  int b = gw / S;
  const float* P = xyz + (size_t)b * N * 3;
  float cx = nxyz[gw*3], cy = nxyz[gw*3+1], cz = nxyz[gw*3+2];
  int* out = idx + (size_t)gw * NSAMP;
  int cnt = 0, firstIdx = 0;
  for (int base = 0; base < N; base += 32) {
    int n = base + lane;
    bool hit = false;
    if (n < N) {
      float dx = P[n*3]-cx, dy = P[n*3+1]-cy, dz = P[n*3+2]-cz;
      hit = (dx*dx + dy*dy + dz*dz) < r2;
    }
    unsigned mask = (unsigned)__ballot(hit);
    if (cnt == 0 && mask) firstIdx = base + __ffs((int)mask) - 1;
    if (hit) {
      int slot = cnt + __popc(mask & ((1u << lane) - 1u));
      if (slot < NSAMP) out[slot] = n;
    }
    cnt += __popc(mask);
    if (cnt >= NSAMP) break;
  }
  if (cnt < NSAMP)
    for (int j = cnt + lane; j < NSAMP; j += 32) out[j] = firstIdx;
}

// ---------------------------------------------------------------- WMMA f32 16x16x4 GEMM on LDS tiles
__device__ __forceinline__ void gemm_lds(const float* src, int pitchS, float* dst, int pitchD,
                                         const float* __restrict__ W, const float* __restrict__ sc,
                                         const float* __restrict__ sh,
                                         int M, int Ncol, int K, int lane, int wave, int nwaves) {
  const int tilesM = M >> 4, tilesN = Ncol >> 4, tiles = tilesM * tilesN;
  const int l15 = lane & 15;
  const int hi2 = (lane >> 4) << 1;           // 0 or 2
  for (int t = wave; t < tiles; t += nwaves) {
    const int tm = t % tilesM, tn = t / tilesM;
    v8f acc = {0.f, 0.f, 0.f, 0.f, 0.f, 0.f, 0.f, 0.f};
    const float* ap = src + (tm * 16 + l15) * pitchS + hi2;   // A: 16x4, lane=M, half=K-pair
    const float* bp = W + (size_t)(tn * 16 + l15) * K + hi2;  // B: 4x16, lane=N, half=K-pair
    for (int k = 0; k < K; k += 4) {
      v2f a = *(const v2f*)(ap + k);
      v2f bb = *(const v2f*)(bp + k);
      acc = __builtin_amdgcn_wmma_f32_16x16x4_f32(false, a, false, bb, (short)0, acc, false, false);
    }
    const int col = tn * 16 + l15;
    const int rbase = tm * 16 + ((lane >> 4) << 3);
    const float s1 = sc[col], s2 = sh[col];
#pragma unroll
    for (int r = 0; r < 8; ++r)
      dst[(rbase + r) * pitchD + col] = fmaxf(acc[r] * s1 + s2, 0.f);   // BN + ReLU epilogue
  }
}

// ---------------------------------------------------------------- fused group -> MLP x3 -> maxpool
template <int NS, int KPAD, int CF, int C1, int C2, int C3>
__global__ void __launch_bounds__(128)
k_sa_stage(const float* __restrict__ xyz, int N,
           const float* __restrict__ feat, int featBS,
           const float* __restrict__ nxyz, const int* __restrict__ idx, int S, float inv_r,
           const float* __restrict__ Wp1, const float* __restrict__ sc1, const float* __restrict__ sh1,
           const float* __restrict__ Wp2, const float* __restrict__ sc2, const float* __restrict__ sh2,
           const float* __restrict__ Wp3, const float* __restrict__ sc3, const float* __restrict__ sh3,
           float* __restrict__ Fout) {
  constexpr int PA = (KPAD > C2 ? KPAD : C2) + 2;   // pitch mod 64 in {2,6}: bank-conflict free
  constexpr int PB = (C1 > C3 ? C1 : C3) + 2;
  __shared__ float bufA[NS * PA];
  __shared__ float bufB[NS * PB];

  const int g = blockIdx.x;             // b*S + s
  const int b = g / S, s = g - b * S;
  const int tid = threadIdx.x;
  const int lane = tid & 31, wave = tid >> 5;
  const int* gi = idx + (size_t)g * NS;
  const float cx = nxyz[g*3], cy = nxyz[g*3+1], cz = nxyz[g*3+2];

  for (int e = tid; e < NS * KPAD; e += 128) {        // build grouped input in LDS
    int j = e / KPAD, c = e - j * KPAD;
    int n = gi[j];
    float v = 0.f;
    if (c < 3) {
      float p = xyz[((size_t)b * N + n) * 3 + c];
      v = (p - (c == 0 ? cx : (c == 1 ? cy : cz))) * inv_r;
    } else if (c < 3 + CF) {
      v = feat[(size_t)b * featBS + (size_t)(c - 3) * N + n];
    }
    bufA[j * PA + c] = v;
  }
  __syncthreads();
  gemm_lds(bufA, PA, bufB, PB, Wp1, sc1, sh1, NS, C1, KPAD, lane, wave, 4);
  __syncthreads();
  gemm_lds(bufB, PB, bufA, PA, Wp2, sc2, sh2, NS, C2, C1, lane, wave, 4);
  __syncthreads();
  gemm_lds(bufA, PA, bufB, PB, Wp3, sc3, sh3, NS, C3, C2, lane, wave, 4);
  __syncthreads();
  for (int c = tid; c < C3; c += 128) {               // max-pool over the group
    float m = bufB[c];
    for (int j = 1; j < NS; ++j) m = fmaxf(m, bufB[j * PB + c]);
    Fout[((size_t)b * C3 + c) * S + s] = m;           // [B][C][S] channel-major
  }
}

// ---------------------------------------------------------------- misc
__global__ void k_track(const int* __restrict__ oldp, const int* __restrict__ inds,
                        int* __restrict__ out, int Sprev, int S, int n) {
  int g = blockIdx.x * blockDim.x + threadIdx.x;
  if (g >= n) return;
  int b = g / S;
  out[g] = oldp[(size_t)b * Sprev + inds[g]];
}
__global__ void k_copy_f(const float* __restrict__ s, float* __restrict__ d, int n) {
  int i = blockIdx.x * blockDim.x + threadIdx.x;
  if (i < n) d[i] = s[i];
}
__global__ void k_i2f(const int* __restrict__ s, float* __restrict__ d, int n) {
  int i = blockIdx.x * blockDim.x + threadIdx.x;
  if (i < n) d[i] = (float)s[i];
}

static inline int cdiv(int a, int b) { return (a + b - 1) / b; }

extern "C" void kernel_launch(void* const* d_in, const int* in_sizes, int n_in,
                              void* d_out, int out_size, void* d_ws, size_t ws_size,
                              hipStream_t stream) {
  const float* input = (const float*)d_in[0];
  // params flattened: 3 stages x 3 layers x (W, bias, gamma, beta)
  const float* P[36];
  for (int i = 0; i < 36 && i + 1 < n_in; ++i) P[i] = (const float*)d_in[i + 1];

  const int S1 = 2048, S2 = 1024, S3 = 512;
  const int ns1 = 64, ns2 = 32, ns3 = 16;
  const int cout1[3] = {64, 64, 128}, cin1[3] = {6, 64, 64},   kp1[3] = {8, 64, 64};
  const int cout2[3] = {128, 128, 256}, cin2[3] = {131, 128, 128}, kp2[3] = {132, 128, 128};
  const int cout3[3] = {256, 256, 256}, cin3[3] = {259, 256, 256}, kp3[3] = {260, 256, 256};

  char* ws = (char*)d_ws;
  size_t off = 0;
  auto alloc = [&](size_t bytes) -> void* {
    off = (off + 255) & ~(size_t)255;
    void* p = ws + off; off += bytes; return p;
  };

  float* xyz0 = (float*)alloc((size_t)BQ * N0 * 3 * 4);
  float* nxyz1 = (float*)alloc((size_t)BQ * S1 * 3 * 4);
  float* nxyz2 = (float*)alloc((size_t)BQ * S2 * 3 * 4);
  float* nxyz3 = (float*)alloc((size_t)BQ * S3 * 3 * 4);
  int* inds1 = (int*)alloc((size_t)BQ * S1 * 4);
  int* inds2 = (int*)alloc((size_t)BQ * S2 * 4);
  int* inds3 = (int*)alloc((size_t)BQ * S3 * 4);
  int* idx1 = (int*)alloc((size_t)BQ * S1 * ns1 * 4);
  int* idx2 = (int*)alloc((size_t)BQ * S2 * ns2 * 4);
  int* idx3 = (int*)alloc((size_t)BQ * S3 * ns3 * 4);
  int* old2 = (int*)alloc((size_t)BQ * S2 * 4);
  int* old3 = (int*)alloc((size_t)BQ * S3 * 4);
  float* F1 = (float*)alloc((size_t)BQ * 128 * S1 * 4);
  float* F2 = (float*)alloc((size_t)BQ * 256 * S2 * 4);

  float *Wp[9], *sc[9], *sh[9];
  const int* allco[3] = {cout1, cout2, cout3};
  const int* allci[3] = {cin1, cin2, cin3};
  const int* allkp[3] = {kp1, kp2, kp3};
  for (int st = 0; st < 3; ++st)
    for (int l = 0; l < 3; ++l) {
      int li = st * 3 + l;
      Wp[li] = (float*)alloc((size_t)allco[st][l] * allkp[st][l] * 4);
      sc[li] = (float*)alloc((size_t)allco[st][l] * 4);
      sh[li] = (float*)alloc((size_t)allco[st][l] * 4);
    }

  // -------- prepack weights / fold BN
  for (int st = 0; st < 3; ++st)
    for (int l = 0; l < 3; ++l) {
      int li = st * 3 + l;
      const float* w = P[li * 4 + 0]; const float* bi = P[li * 4 + 1];
      const float* ga = P[li * 4 + 2]; const float* be = P[li * 4 + 3];
      int n = allco[st][l] * allkp[st][l];
      k_pack_w<<<cdiv(n, 256), 256, 0, stream>>>(w, bi, ga, be, Wp[li], sc[li], sh[li],
                                                 allco[st][l], allci[st][l], allkp[st][l]);
    }

  // -------- stage 1
  k_xyz0<<<cdiv(BQ * N0 * 3, 256), 256, 0, stream>>>(input, xyz0, BQ * N0 * 3);
  k_fps<<<BQ, FPS_T, 0, stream>>>(xyz0, N0, S1, inds1);
  k_gather_xyz<<<cdiv(BQ * S1 * 3, 256), 256, 0, stream>>>(xyz0, inds1, nxyz1, N0, S1, BQ * S1 * 3);
  k_ballquery<<<cdiv(BQ * S1 * 32, 256), 256, 0, stream>>>(xyz0, nxyz1, N0, S1, ns1, 0.2f * 0.2f, idx1, BQ * S1);
  k_sa_stage<64, 8, 3, 64, 64, 128><<<BQ * S1, 128, 0, stream>>>(
      xyz0, N0, input + 3 * N0, 6 * N0, nxyz1, idx1, S1, 1.f / 0.2f,
      Wp[0], sc[0], sh[0], Wp[1], sc[1], sh[1], Wp[2], sc[2], sh[2], F1);

  // -------- stage 2
  k_fps<<<BQ, FPS_T, 0, stream>>>(nxyz1, S1, S2, inds2);
  k_gather_xyz<<<cdiv(BQ * S2 * 3, 256), 256, 0, stream>>>(nxyz1, inds2, nxyz2, S1, S2, BQ * S2 * 3);
  k_ballquery<<<cdiv(BQ * S2 * 32, 256), 256, 0, stream>>>(nxyz1, nxyz2, S1, S2, ns2, 0.4f * 0.4f, idx2, BQ * S2);
  k_sa_stage<32, 132, 128, 128, 128, 256><<<BQ * S2, 128, 0, stream>>>(
      nxyz1, S1, F1, 128 * S1, nxyz2, idx2, S2, 1.f / 0.4f,
      Wp[3], sc[3], sh[3], Wp[4], sc[4], sh[4], Wp[5], sc[5], sh[5], F2);
  k_track<<<cdiv(BQ * S2, 256), 256, 0, stream>>>(inds1, inds2, old2, S1, S2, BQ * S2);

  // -------- stage 3 (features written straight into d_out)
  float* out = (float*)d_out;
  const int o_fxyz = 0;                       // final xyz           [4,512,3]   6144
  const int o_feat = 6144;                    // features            [4,256,512] 524288
  const int o_xyz1 = 530432, o_ind1 = 555008; // [4,2048,3] / [4,2048]
  const int o_xyz2 = 563200, o_ind2 = 575488; // [4,1024,3] / [4,1024]
  const int o_xyz3 = 579584, o_ind3 = 585728; // [4,512,3]  / [4,512]

  k_fps<<<BQ, FPS_T, 0, stream>>>(nxyz2, S2, S3, inds3);
  k_gather_xyz<<<cdiv(BQ * S3 * 3, 256), 256, 0, stream>>>(nxyz2, inds3, nxyz3, S2, S3, BQ * S3 * 3);
  k_ballquery<<<cdiv(BQ * S3 * 32, 256), 256, 0, stream>>>(nxyz2, nxyz3, S2, S3, ns3, 0.8f * 0.8f, idx3, BQ * S3);
  k_sa_stage<16, 260, 256, 256, 256, 256><<<BQ * S3, 128, 0, stream>>>(
      nxyz2, S2, F2, 256 * S2, nxyz3, idx3, S3, 1.f / 0.8f,
      Wp[6], sc[6], sh[6], Wp[7], sc[7], sh[7], Wp[8], sc[8], sh[8], out + o_feat);
  k_track<<<cdiv(BQ * S3, 256), 256, 0, stream>>>(old2, inds3, old3, S2, S3, BQ * S3);

  // -------- assemble outputs
  k_copy_f<<<cdiv(BQ * S3 * 3, 256), 256, 0, stream>>>(nxyz3, out + o_fxyz, BQ * S3 * 3);
  k_copy_f<<<cdiv(BQ * S1 * 3, 256), 256, 0, stream>>>(nxyz1, out + o_xyz1, BQ * S1 * 3);
  k_i2f<<<cdiv(BQ * S1, 256), 256, 0, stream>>>(inds1, out + o_ind1, BQ * S1);
  k_copy_f<<<cdiv(BQ * S2 * 3, 256), 256, 0, stream>>>(nxyz2, out + o_xyz2, BQ * S2 * 3);
  k_i2f<<<cdiv(BQ * S2, 256), 256, 0, stream>>>(old2, out + o_ind2, BQ * S2);
  k_copy_f<<<cdiv(BQ * S3 * 3, 256), 256, 0, stream>>>(nxyz3, out + o_xyz3, BQ * S3 * 3);
  k_i2f<<<cdiv(BQ * S3, 256), 256, 0, stream>>>(old3, out + o_ind3, BQ * S3);
}